// LGCN_28166395527749
// MI455X (gfx1250) — compile-verified
//
#include <hip/hip_runtime.h>

#define DFEAT 64
#define KHOPS 8
#define OUTD ((KHOPS + 1) * DFEAT)   // 576
#define CHUNK 256                    // edges staged in LDS per block

typedef unsigned int u32;
typedef unsigned long long u64;
typedef u32 v4u __attribute__((ext_vector_type(4)));
typedef int  v4i __attribute__((ext_vector_type(4)));
typedef int  v8i __attribute__((ext_vector_type(8)));

// ---------------------------------------------------------------------------
// TDM: 1-D (1 row x nelem) async DMA of 4-byte elements global -> LDS.
// Descriptor per CDNA5 ISA ch.8: group0 = {count, lds_addr, global_addr, type=2},
// group1 = {data_size=4B, tensor_dim0=nelem, tensor_dim1=1, tile_dim0, tile_dim1=1}.
// Issued once per wave (EXEC ignored); tracked by TENSORcnt.
// Toolchain builtin is the 6-arg form: (v4u, v8i, v4i, v4i, v8i, i32 cpol).
// ---------------------------------------------------------------------------
__device__ __forceinline__ void tdm_load_1d(u64 gaddr, u32 lds_addr, u32 nelem) {
    v4u g0;
    g0.x = 1u;                                          // count=1, user descriptor
    g0.y = lds_addr;                                    // LDS byte address
    g0.z = (u32)(gaddr & 0xFFFFFFFFull);                // global_addr[31:0]
    g0.w = (u32)((gaddr >> 32) & 0x01FFFFFFull)         // global_addr[56:32]
         | (2u << 30);                                  // type = 2 ("image")
    v8i g1;
    g1[0] = (int)(2u << 16);                            // data_size=4B, no multicast/pad
    g1[1] = (int)((nelem & 0xFFFFu) << 16);             // tensor_dim0[15:0]
    g1[2] = (int)(((nelem >> 16) & 0xFFFFu) | (1u << 16)); // tensor_dim0[31:16], tensor_dim1=1
    g1[3] = (int)((nelem & 0xFFFFu) << 16);             // tile_dim0 = nelem
    g1[4] = 1;                                          // tile_dim1 = 1, tile_dim2 = 0
    g1[5] = (int)nelem;                                 // tensor_dim0_stride
    g1[6] = 0;
    g1[7] = 0;
    v4i z4 = {0, 0, 0, 0};
    v8i z8 = {0, 0, 0, 0, 0, 0, 0, 0};
    __builtin_amdgcn_tensor_load_to_lds(g0, g1, z4, z4, z8, 0);
}

// --------------------------- degree / normalization ------------------------

__global__ void k_zero_deg(int* __restrict__ deg, int n) {
    int i = blockIdx.x * blockDim.x + threadIdx.x;
    if (i < n) deg[i] = 0;
}

__global__ void k_count_deg(const int* __restrict__ src, const int* __restrict__ dst,
                            int* __restrict__ deg, int e) {
    int i = blockIdx.x * blockDim.x + threadIdx.x;
    if (i < e) {
        int s = src[i];
        if (s != dst[i]) atomicAdd(&deg[s], 1);   // self-loop edges get weight 0
    }
}

__global__ void k_dinv(const int* __restrict__ deg, float* __restrict__ dinv, int n) {
    int i = blockIdx.x * blockDim.x + threadIdx.x;
    if (i < n) dinv[i] = rsqrtf((float)(deg[i] + 1));  // +1 = appended self loop; deg>0 always
}

__global__ void k_norm(const int* __restrict__ src, const int* __restrict__ dst,
                       const float* __restrict__ dinv, float* __restrict__ nrm, int e) {
    int i = blockIdx.x * blockDim.x + threadIdx.x;
    if (i < e) {
        int s = src[i], d = dst[i];
        nrm[i] = (s == d) ? 0.0f : dinv[s] * dinv[d];
    }
}

// --------------------------- hop-0 copy ------------------------------------
// out[:, 0:64] = X ; 16 threads per node, float4 (256B/row coalesced)
__global__ void k_copy_x(const float* __restrict__ x, float* __restrict__ out, int n) {
    int t = blockIdx.x * blockDim.x + threadIdx.x;
    if (t < n * 16) {
        int i = t >> 4, q = t & 15;
        float4 v = ((const float4*)(x + (size_t)i * DFEAT))[q];
        ((float4*)(out + (size_t)i * OUTD))[q] = v;
    }
}

// --------------------------- self-loop init --------------------------------
// out[i, k-block] = dinv[i]^2 * out[i, (k-1)-block]
__global__ void k_self_loop(const float* __restrict__ dinv, float* __restrict__ out,
                            int n, int kin_off, int kout_off) {
    int t = blockIdx.x * blockDim.x + threadIdx.x;
    if (t < n * 16) {
        int i = t >> 4, q = t & 15;
        float di = dinv[i];
        float s = di * di;
        float4 v = ((const float4*)(out + (size_t)i * OUTD + kin_off))[q];
        float4 r = {s * v.x, s * v.y, s * v.z, s * v.w};
        ((float4*)(out + (size_t)i * OUTD + kout_off))[q] = r;
    }
}

// --------------------------- edge scatter ----------------------------------
// One block stages CHUNK edges' (src,dst,norm) into LDS via TDM, then each
// wave32 processes one edge at a time: 32 lanes x float2 = 64-feature row.
// out[dst, k] += norm * out[src, k-1]  via global_atomic_add_f32 (L2-resident).
__global__ void __launch_bounds__(256) k_scatter(
        const int* __restrict__ src, const int* __restrict__ dst,
        const float* __restrict__ nrm, float* __restrict__ out,
        int e, int kin_off, int kout_off) {
    __shared__ __align__(16) int   s_src[CHUNK];
    __shared__ __align__(16) int   s_dst[CHUNK];
    __shared__ __align__(16) float s_nrm[CHUNK];

    int base = blockIdx.x * CHUNK;
    int cnt = e - base;
    if (cnt > CHUNK) cnt = CHUNK;
    if (cnt <= 0) return;

    if ((threadIdx.x >> 5) == 0) {   // wave 0 issues the three DMAs
        tdm_load_1d((u64)(uintptr_t)(src + base), (u32)(uintptr_t)&s_src[0], (u32)cnt);
        tdm_load_1d((u64)(uintptr_t)(dst + base), (u32)(uintptr_t)&s_dst[0], (u32)cnt);
        tdm_load_1d((u64)(uintptr_t)(nrm + base), (u32)(uintptr_t)&s_nrm[0], (u32)cnt);
        __builtin_amdgcn_s_wait_tensorcnt(0);
    }
    __syncthreads();

    int lane = threadIdx.x & 31;
    int wid  = threadIdx.x >> 5;
    for (int i = wid; i < cnt; i += 8) {
        float w = s_nrm[i];
        if (w == 0.0f) continue;     // self-loop edge, weight zeroed
        int s = s_src[i], d = s_dst[i];
        float2 v = ((const float2*)(out + (size_t)s * OUTD + kin_off))[lane];
        float* po = out + (size_t)d * OUTD + kout_off + lane * 2;
        unsafeAtomicAdd(po,     w * v.x);
        unsafeAtomicAdd(po + 1, w * v.y);
    }
}

// ---------------------------------------------------------------------------

extern "C" void kernel_launch(void* const* d_in, const int* in_sizes, int n_in,
                              void* d_out, int out_size, void* d_ws, size_t ws_size,
                              hipStream_t stream) {
    const float* feat = (const float*)d_in[0];
    const int*   ei   = (const int*)d_in[1];
    int n = in_sizes[0] / DFEAT;   // 50000
    int e = in_sizes[1] / 2;       // 800000
    const int* src = ei;
    const int* dst = ei + e;
    float* out = (float*)d_out;

    // workspace: deg (int n) | dinv (float n) | norm (float e), 256B-aligned
    char* ws = (char*)d_ws;
    size_t off = 0;
    int*   deg  = (int*)(ws + off);  off += ((size_t)n * 4 + 255) & ~(size_t)255;
    float* dinv = (float*)(ws + off); off += ((size_t)n * 4 + 255) & ~(size_t)255;
    float* nrm  = (float*)(ws + off);

    int tb = 256;
    int gN  = (n + tb - 1) / tb;
    int gE  = (e + tb - 1) / tb;
    int gNf = (n * 16 + tb - 1) / tb;   // 16 threads/node (float4)
    int gC  = (e + CHUNK - 1) / CHUNK;

    k_zero_deg<<<gN, tb, 0, stream>>>(deg, n);
    k_count_deg<<<gE, tb, 0, stream>>>(src, dst, deg, e);
    k_dinv<<<gN, tb, 0, stream>>>(deg, dinv, n);
    k_norm<<<gE, tb, 0, stream>>>(src, dst, dinv, nrm, e);
    k_copy_x<<<gNf, tb, 0, stream>>>(feat, out, n);

    for (int k = 1; k <= KHOPS; ++k) {
        int kin  = (k - 1) * DFEAT;
        int kout = k * DFEAT;
        k_self_loop<<<gNf, tb, 0, stream>>>(dinv, out, n, kin, kout);
        k_scatter<<<gC, tb, 0, stream>>>(src, dst, nrm, out, e, kin, kout);
    }
}